// MinimalThinkingRefiner_50525995270582
// MI455X (gfx1250) — compile-verified
//
#include <hip/hip_runtime.h>

// MinimalThinkingRefiner — streaming elementwise op, bandwidth bound.
// Roofline: 256 MB in + 256 MB out = 512 MB @ 23.3 TB/s ≈ 22 us; ~0.26 GFLOP.
// WMMA is irrelevant (no GEMM structure). Optimal CDNA5 shape:
//   - b128 global loads/stores, non-temporal TH on both 256 MB streams
//     (streams exceed the 192 MB L2 and are never reused; NT keeps the
//      16384x-reused scale/shift/mask resident instead)
//   - wave-uniform per-row mask branch (scalar s_cmp/s_cbranch, no exec mask)
//   - compile-time trip count: 4 fully unrolled b128 loads per thread with
//     immediate offsets -> clause-able, 4 loads in flight per wave.

typedef __attribute__((ext_vector_type(4))) float float4v;

template <int H4C>  // H/4 in float4 units; H4C % 256 == 0
__global__ __launch_bounds__(256) void
refiner_fixed(const float* __restrict__ x,     // [rows, H]
              const int*   __restrict__ mask,  // [rows]
              const float* __restrict__ scale, // [H]
              const float* __restrict__ shift, // [H]
              const float* __restrict__ alpha_p,
              float*       __restrict__ out)   // [rows, H]
{
    constexpr int ITER = H4C / 256;
    const int row = blockIdx.x;
    const long long rowBase = (long long)row * ((long long)H4C * 4);

    const float4v* __restrict__ xin  = (const float4v*)(x + rowBase);
    float4v*       __restrict__ yout = (float4v*)(out + rowBase);
    const float4v* __restrict__ sc4  = (const float4v*)scale;
    const float4v* __restrict__ sh4  = (const float4v*)shift;

    const int tid = threadIdx.x;
    // Uniform per-row predicate: scalar load + scalar branch for the wave.
    const bool thinking = (mask[row] == 2);

    if (thinking) {
        const float alpha = *alpha_p;

        float4v xv[ITER], sv[ITER], tv[ITER], r[ITER];
        // All streaming loads first: one base address, immediate offsets
        // 0 / 4096 / 8192 / 12288 bytes -> clause of 4 global_load_b128 NT.
#pragma unroll
        for (int k = 0; k < ITER; ++k)
            xv[k] = __builtin_nontemporal_load(&xin[tid + 256 * k]);
        // scale/shift: hot in WGP$/L2 (32 KB, reused by every row) -> regular TH.
#pragma unroll
        for (int k = 0; k < ITER; ++k) {
            sv[k] = sc4[tid + 256 * k];
            tv[k] = sh4[tid + 256 * k];
        }
#pragma unroll
        for (int k = 0; k < ITER; ++k) {
#pragma unroll
            for (int j = 0; j < 4; ++j) {
                float refined = __builtin_fmaf(xv[k][j], sv[k][j], tv[k][j]);
                r[k][j] = __builtin_fmaf(alpha, refined, xv[k][j]);
            }
        }
        // All streaming stores last: clause of 4 global_store_b128 NT.
#pragma unroll
        for (int k = 0; k < ITER; ++k)
            __builtin_nontemporal_store(r[k], &yout[tid + 256 * k]);
    } else {
        // Pass-through row: pure b128 NT copy, loads first then stores.
        float4v xv[ITER];
#pragma unroll
        for (int k = 0; k < ITER; ++k)
            xv[k] = __builtin_nontemporal_load(&xin[tid + 256 * k]);
#pragma unroll
        for (int k = 0; k < ITER; ++k)
            __builtin_nontemporal_store(xv[k], &yout[tid + 256 * k]);
    }
}

// Generic fallback for shapes other than H==4096 (not used by this harness).
__global__ __launch_bounds__(256) void
refiner_generic(const float* __restrict__ x, const int* __restrict__ mask,
                const float* __restrict__ scale, const float* __restrict__ shift,
                const float* __restrict__ alpha_p, float* __restrict__ out, int H4)
{
    const int row = blockIdx.x;
    const long long rowBase = (long long)row * ((long long)H4 * 4);
    const float4v* __restrict__ xin  = (const float4v*)(x + rowBase);
    float4v*       __restrict__ yout = (float4v*)(out + rowBase);
    const float4v* __restrict__ sc4  = (const float4v*)scale;
    const float4v* __restrict__ sh4  = (const float4v*)shift;
    const bool thinking = (mask[row] == 2);
    if (thinking) {
        const float alpha = *alpha_p;
        for (int i = threadIdx.x; i < H4; i += 256) {
            float4v xv = __builtin_nontemporal_load(&xin[i]);
            float4v sv = sc4[i], tv = sh4[i], r;
#pragma unroll
            for (int j = 0; j < 4; ++j) {
                float refined = __builtin_fmaf(xv[j], sv[j], tv[j]);
                r[j] = __builtin_fmaf(alpha, refined, xv[j]);
            }
            __builtin_nontemporal_store(r, &yout[i]);
        }
    } else {
        for (int i = threadIdx.x; i < H4; i += 256) {
            float4v xv = __builtin_nontemporal_load(&xin[i]);
            __builtin_nontemporal_store(xv, &yout[i]);
        }
    }
}

extern "C" void kernel_launch(void* const* d_in, const int* in_sizes, int n_in,
                              void* d_out, int out_size, void* d_ws, size_t ws_size,
                              hipStream_t stream) {
    // setup_inputs order: hidden_states, input_mask, scale, shift, alpha
    const float* x     = (const float*)d_in[0];
    const int*   mask  = (const int*)d_in[1];
    const float* scale = (const float*)d_in[2];
    const float* shift = (const float*)d_in[3];
    const float* alpha = (const float*)d_in[4];
    float*       out   = (float*)d_out;

    const int rows = in_sizes[1];   // B*S = 16384
    const int H    = in_sizes[2];   // 4096
    const int H4   = H / 4;

    if (H4 == 1024) {
        // One block per row: 256 threads (8 wave32), 4 unrolled float4 each.
        refiner_fixed<1024><<<rows, 256, 0, stream>>>(x, mask, scale, shift,
                                                      alpha, out);
    } else {
        refiner_generic<<<rows, 256, 0, stream>>>(x, mask, scale, shift,
                                                  alpha, out, H4);
    }
    (void)d_ws; (void)ws_size; (void)n_in; (void)out_size;
}